// WeightInitializerNorm_75711683494079
// MI455X (gfx1250) — compile-verified
//
#include <hip/hip_runtime.h>
#include <stdint.h>

// Problem constants (match reference).
#define T_TASKS 256
#define S_SUP   512
#define D_FEAT  640
#define W_WAY   64

#define CHUNK_ROWS 16                       // support rows staged per TDM op
#define N_CHUNKS   (S_SUP / CHUNK_ROWS)     // 32

typedef __attribute__((ext_vector_type(2)))  float    v2f;
typedef __attribute__((ext_vector_type(8)))  float    v8f;
typedef __attribute__((ext_vector_type(4)))  uint32_t v4u;
typedef __attribute__((ext_vector_type(4)))  int      v4i;
typedef __attribute__((ext_vector_type(8)))  int      v8i;

// Dynamic LDS layout:
//   pos    [W][D]           f32 : 163840 B  (per-class feature sums, accumulated)
//   total  [D]              f32 :   2560 B
//   lab    [S]              i32 :   2048 B
//   kf/fgs/bgs/dens [W] x4  f32 :   1024 B
//   stage  [2][CHUNK][D]    f32 :  81920 B  (TDM double buffer)
static constexpr size_t LDS_STATS = (size_t)W_WAY * D_FEAT * 4 + D_FEAT * 4 +
                                    S_SUP * 4 + 4 * W_WAY * 4;          // 169472
static constexpr size_t LDS_BYTES = LDS_STATS +
                                    2 * (size_t)CHUNK_ROWS * D_FEAT * 4; // 251392

// Issue one TDM tensor_load_to_lds: copy CHUNK_ROWS x D_FEAT f32 tile
// (contiguous rows, stride D_FEAT) from global 'ga' into LDS byte offset 'lds_off'.
__device__ __forceinline__ void tdm_load_chunk(uint64_t ga, uint32_t lds_off) {
    v4u g0;
    g0[0] = 1u;                                   // [1:0]  count = 1 (user mode)
    g0[1] = lds_off;                              // [63:32] lds_addr (bytes)
    g0[2] = (uint32_t)(ga & 0xFFFFFFFFu);         // [95:64] global_addr lo
    g0[3] = ((uint32_t)(ga >> 32) & 0x01FFFFFFu)  // [120:96] global_addr hi
            | 0x80000000u;                        // [127:126] type = 2 ("image")
    v8i g1;
    g1[0] = 0x00020000;                           // data_size = 2 (4 bytes/elem)
    g1[1] = (int)((uint32_t)D_FEAT << 16);        // tensor_dim0[15:0] in [63:48]
    g1[2] = (int)((uint32_t)CHUNK_ROWS << 16);    // dim0 hi = 0; tensor_dim1 = 16
    g1[3] = (int)((uint32_t)D_FEAT << 16);        // dim1 hi = 0; tile_dim0 = 640
    g1[4] = CHUNK_ROWS;                           // tile_dim1 = 16; tile_dim2 = 0
    g1[5] = D_FEAT;                               // tensor_dim0_stride lo = 640
    g1[6] = 0;                                    // stride0 hi; dim1_stride lo (unused, 2D)
    g1[7] = 0;
    v4i gz = {0, 0, 0, 0};
#if defined(__clang_major__) && (__clang_major__ >= 23)
    v8i gz8 = {0, 0, 0, 0, 0, 0, 0, 0};
    __builtin_amdgcn_tensor_load_to_lds(g0, g1, gz, gz, gz8, 0);
#else
    __builtin_amdgcn_tensor_load_to_lds(g0, g1, gz, gz, 0);
#endif
}

__global__ __launch_bounds__(256)
void proto_weight_init_kernel(const float* __restrict__ support,
                              const int*   __restrict__ labels,
                              const float* __restrict__ target_pos,
                              const float* __restrict__ target_neg,
                              float*       __restrict__ out) {
    extern __shared__ char smem_raw[];
    float* pos    = (float*)smem_raw;                // [W][D]
    float* total  = pos + W_WAY * D_FEAT;            // [D]
    int*   lab    = (int*)(total + D_FEAT);          // [S]
    float* kf     = (float*)(lab + S_SUP);           // [W]
    float* fgs    = kf  + W_WAY;                     // [W]
    float* bgs    = fgs + W_WAY;                     // [W]
    float* dens   = bgs + W_WAY;                     // [W]
    float* stage0 = dens + W_WAY;                    // [CHUNK][D]
    float* stage1 = stage0 + CHUNK_ROWS * D_FEAT;    // [CHUNK][D]

    const int t    = blockIdx.x;
    const int tid  = threadIdx.x;
    const int lane = tid & 31;
    const int wave = tid >> 5;     // 0..7
    const int half = lane >> 4;    // lane group 0/1
    const int l16  = lane & 15;

    const float* sup_t = support + (size_t)t * S_SUP * D_FEAT;

    // LDS byte offsets of the staging buffers (low 32 bits of the flat address
    // of an LDS pointer are the in-LDS byte offset on CDNA5).
    const uint32_t stage_off0 = (uint32_t)(size_t)stage0;
    const uint32_t stage_off1 = (uint32_t)(size_t)stage1;

    // ---- Prologue: kick off chunk 0 DMA, stage labels meanwhile ----
    if (wave == 0 && lane == 0)
        tdm_load_chunk((uint64_t)(uintptr_t)sup_t, stage_off0);
    for (int i = tid; i < S_SUP; i += 256)
        lab[i] = labels[(size_t)t * S_SUP + i];

    // ---- Phase 1: pos[w][d] = sum_s onehot[w,s]*support[s,d] via fp32 WMMA,
    //      support streamed HBM -> LDS by the Tensor Data Mover (double buffer).
    // A (16x4 f32): lanes 0-15 hold M=lane; VGPR e holds K = half*2 + e.
    // B (4x16 f32): lanes hold N=lane&15;  VGPR e holds K = half*2 + e.
    // C/D (16x16 f32): VGPR r -> M = r + 8*half, N = lane&15.
    for (int c = 0; c < N_CHUNKS; ++c) {
        __syncthreads();   // everyone done reading the buffer we are about to refill
        if (wave == 0 && lane == 0) {
            if (c + 1 < N_CHUNKS) {
                const uint64_t ga = (uint64_t)(uintptr_t)
                    (sup_t + (size_t)(c + 1) * CHUNK_ROWS * D_FEAT);
                tdm_load_chunk(ga, ((c + 1) & 1) ? stage_off1 : stage_off0);
                __builtin_amdgcn_s_wait_tensorcnt(1);  // chunk c landed
            } else {
                __builtin_amdgcn_s_wait_tensorcnt(0);  // last chunk landed
            }
        }
        __syncthreads();   // chunk c visible to all waves
        const float* buf = (c & 1) ? stage1 : stage0;
        const int sc = c * CHUNK_ROWS;

        for (int tile = wave; tile < (W_WAY / 16) * (D_FEAT / 16); tile += 8) {
            const int w0   = (tile & 3) * 16;
            const int d0   = (tile >> 2) * 16;
            const int m    = w0 + l16;
            const int wrow = w0 + 8 * half;
            v8f acc;
            if (c == 0) {
                acc = (v8f){};
            } else {
#pragma unroll
                for (int r = 0; r < 8; ++r)
                    acc[r] = pos[(wrow + r) * D_FEAT + d0 + l16];
            }
#pragma unroll
            for (int kk = 0; kk < CHUNK_ROWS; kk += 4) {
                const int kg = sc + kk + half * 2;   // global support row (A index)
                const int kl = kk + half * 2;        // row inside staging buffer
                v2f a, b;
                a.x = (lab[kg]     == m) ? 1.0f : 0.0f;
                a.y = (lab[kg + 1] == m) ? 1.0f : 0.0f;
                const float* bp = buf + kl * D_FEAT + d0 + l16;
                b.x = bp[0];
                b.y = bp[D_FEAT];
                acc = __builtin_amdgcn_wmma_f32_16x16x4_f32(
                    /*neg_a=*/false, a, /*neg_b=*/false, b,
                    /*c_mod=*/(short)0, acc, /*reuse_a=*/false, /*reuse_b=*/false);
            }
#pragma unroll
            for (int r = 0; r < 8; ++r)
                pos[(wrow + r) * D_FEAT + d0 + l16] = acc[r];
        }
    }
    __syncthreads();

    // ---- Phase 2: k_shot histogram + per-dim totals (all from LDS) ----
    if (tid < W_WAY) {
        int cnt = 0;
        for (int s = 0; s < S_SUP; ++s)
            cnt += (lab[s] == tid) ? 1 : 0;
        kf[tid] = (float)cnt;
    }
    for (int d = tid; d < D_FEAT; d += 256) {
        float acc = 0.0f;
        for (int w = 0; w < W_WAY; ++w)
            acc += pos[w * D_FEAT + d];
        total[d] = acc;
    }
    __syncthreads();

    // ---- Phase 3: ff/bb/fb per class (wave32 reductions), then scales ----
    const float tp = target_pos[0];
    const float tn = target_neg[0];
    for (int w = wave; w < W_WAY; w += 8) {
        const float k    = kf[w];
        const float rpos = 1.0f / k;
        const float rneg = 1.0f / ((float)S_SUP - k);
        float ff = 0.0f, bb = 0.0f, fb = 0.0f;
        for (int d = lane; d < D_FEAT; d += 32) {
            const float ps = pos[w * D_FEAT + d];
            const float p  = ps * rpos;
            const float n  = (total[d] - ps) * rneg;
            ff += p * p;
            bb += n * n;
            fb += p * n;
        }
#pragma unroll
        for (int off = 16; off > 0; off >>= 1) {
            ff += __shfl_xor(ff, off, 32);
            bb += __shfl_xor(bb, off, 32);
            fb += __shfl_xor(fb, off, 32);
        }
        if (lane == 0) {
            dens[w] = fmaxf(ff * bb - fb * fb, 1e-6f);
            fgs[w]  = tp * bb - tn * fb;
            bgs[w]  = tp * fb - tn * ff;
        }
    }
    __syncthreads();

    // ---- Phase 4: weights = (fg*p - bg*n)/den, coalesced non-temporal store ----
    float* out_t = out + (size_t)t * W_WAY * D_FEAT;
    for (int idx = tid; idx < W_WAY * D_FEAT; idx += 256) {
        const int w = idx / D_FEAT;
        const int d = idx - w * D_FEAT;
        const float ps = pos[idx];
        const float k  = kf[w];
        const float p  = ps / k;
        const float n  = (total[d] - ps) / ((float)S_SUP - k);
        const float v  = (fgs[w] * p - bgs[w] * n) / dens[w];
        __builtin_nontemporal_store(v, &out_t[idx]);
    }
}

extern "C" void kernel_launch(void* const* d_in, const int* in_sizes, int n_in,
                              void* d_out, int out_size, void* d_ws, size_t ws_size,
                              hipStream_t stream) {
    (void)in_sizes; (void)n_in; (void)out_size; (void)d_ws; (void)ws_size;
    const float* support    = (const float*)d_in[0];
    const int*   labels     = (const int*)d_in[1];
    // d_in[2] = n_way (constant 64, baked in)
    const float* target_pos = (const float*)d_in[3];
    const float* target_neg = (const float*)d_in[4];
    float* out = (float*)d_out;

    // Allow > 64KB dynamic LDS (CDNA5 WGP supports up to 320KB per workgroup).
    hipFuncSetAttribute((const void*)proto_weight_init_kernel,
                        hipFuncAttributeMaxDynamicSharedMemorySize,
                        (int)LDS_BYTES);

    dim3 grid(T_TASKS);
    dim3 block(256);
    proto_weight_init_kernel<<<grid, block, LDS_BYTES, stream>>>(
        support, labels, target_pos, target_neg, out);
}